// CompProbModel_44959717655006
// MI455X (gfx1250) — compile-verified
//
#include <hip/hip_runtime.h>
#include <math.h>

// ---- problem constants (from reference) ----
#define NXC 120
#define NYC 55
#define NTC 40
#define PC  22
#define FC  (NXC*NYC)          // 6600
constexpr float G_C    = 10.72468f;
constexpr float REAXT  = 0.2f;
constexpr float A_MAXC = 7.25f;
constexpr float S_MAXC = 9.25f;
constexpr float SIGK   = 3.14f / (1.732f * 0.5f);   // 3.6258...

// T[t] = linspace(0.1,4.0,40) => 0.1*(t+1)
__device__ __forceinline__ float t_of(int t) { return 0.1f * (float)(t + 1); }

// ======================================================================
// Kernel 1: p_int table  [cell][t][player] (fp32, 23.2 MB -> lives in L2)
// one thread per (cell, player); 40 sigmoids each
// ======================================================================
__global__ void pint_kernel(const float* __restrict__ frame,
                            float* __restrict__ table) {
    int id = blockIdx.x * blockDim.x + threadIdx.x;
    if (id >= FC * PC) return;
    int f = id / PC, p = id % PC;
    int ix = f % NXC, iy = f / NXC;
    float fx = 0.5f + (float)ix;
    float fy = (iy == 0) ? -0.2f : (-0.5f + (float)iy);

    const float* pl = frame + p * 14;
    float px = pl[1], py = pl[2], vx = pl[3], vy = pl[4], ax = pl[5], ay = pl[6];
    float vxr = ax * REAXT + vx;
    float vyr = ay * REAXT + vy;
    float xr  = px + vx * REAXT + 0.5f * ax * REAXT * REAXT;
    float yr  = py + vy * REAXT + 0.5f * ay * REAXT * REAXT;

    float dx = fx - xr, dy = fy - yr;
    float dmag = sqrtf(dx * dx + dy * dy);
    float s0 = (dx * vxr + dy * vyr) / dmag;
    s0 = fminf(fmaxf(s0, -S_MAXC), S_MAXC);
    float t_lt = (S_MAXC - s0) / A_MAXC;
    float d_lt = t_lt * (s0 + S_MAXC) * 0.5f;
    if (d_lt > dmag) {
        float q = s0 / A_MAXC;
        t_lt = -q + sqrtf(q * q + 2.0f * dmag / A_MAXC);
    }
    d_lt = fmaxf(fminf(d_lt, dmag), 0.0f);
    float t_tot = REAXT + t_lt + (dmag - d_lt) / S_MAXC;

    float* row = table + (size_t)f * NTC * PC + p;
    for (int t = 0; t < NTC; ++t) {
        float arg = SIGK * (t_of(t) - t_tot);
        row[(size_t)t * PC] = 1.0f / (1.0f + expf(-arg));
    }
}

// ======================================================================
// Kernel 2: path walk. One wave32 per (f, jf); lane = player (0..21).
// Per step: coalesced 88B row gather from L2-resident table, butterfly
// reduce, sequential survival product, accumulate -> ppc_ind.
// ======================================================================
__global__ void path_kernel(const float* __restrict__ frame,
                            const float* __restrict__ table,
                            float* __restrict__ ind_out) {
    int gtid = blockIdx.x * blockDim.x + threadIdx.x;
    int wave = gtid >> 5;
    int lane = gtid & 31;
    if (wave >= FC * NTC) return;
    int f = wave / NTC, jf = wave % NTC;

    int ix = f % NXC, iy = f / NXC;
    float fx = 0.5f + (float)ix;
    float fy = (iy == 0) ? -0.2f : (-0.5f + (float)iy);
    float bsx = frame[8], bsy = frame[9];            // ball start (player 0)
    float Tj  = t_of(jf);
    float vxb = (fx - bsx) / Tj;
    float vyb = (fy - bsy) / Tj;
    float vz0 = Tj * (G_C * 0.5f);

    float run = 1.0f, acc = 0.0f;
    for (int it = 0; it <= jf; ++it) {
        float Ti = t_of(it);
        float posx = bsx + vxb * Ti;
        float posy = bsy + vyb * Ti;
        int xi = (int)rintf(fminf(fmaxf(posx, 0.0f), (float)(NXC - 1)));
        int yi = (int)rintf(fminf(fmaxf(posy, 0.0f), (float)(NYC - 1)));
        int cell = yi * NXC + xi;
        float z  = 2.0f + vz0 * Ti - 0.5f * G_C * Ti * Ti;
        float lz = (z < 3.0f && z > 0.0f) ? 1.0f : 0.0f;

        // prefetch next step's row (global_prefetch_b8)
        if (lane == 0 && it < jf) {
            float Tn = t_of(it + 1);
            float nx = bsx + vxb * Tn, ny = bsy + vyb * Tn;
            int xin = (int)rintf(fminf(fmaxf(nx, 0.0f), (float)(NXC - 1)));
            int yin = (int)rintf(fminf(fmaxf(ny, 0.0f), (float)(NYC - 1)));
            __builtin_prefetch(table + ((size_t)(yin * NXC + xin) * NTC + it + 1) * PC, 0, 3);
        }

        float val = 0.0f;
        if (lane < PC) val = table[((size_t)cell * NTC + it) * PC + lane] * lz;

        float s = val;                                // wave32 butterfly sum
        #pragma unroll
        for (int off = 16; off; off >>= 1) s += __shfl_xor(s, off, 32);

        float denom = fmaxf(1.0f, s);
        float pn = val / denom;
        acc += run * pn;                              // shift[it] * p_norm
        run *= (1.0f - s / denom);                    // survival product
    }
    if (lane < PC) ind_out[((size_t)f * NTC + jf) * PC + lane] = acc;
}

// ======================================================================
// Kernel 3: ppc_off/def = ppc_ind(264000x22) x W(22x2) via WMMA f32 16x16x4
// A layout (ISA 7.12.2): lanes 0-15 M, v0=K0/v1=K1; lanes 16-31 K+2.
// B assumed symmetric (v0: K=2*kh, v1: K=2*kh+1, lane%16 = N).
// C/D: vgpr r -> M=r (lanes 0-15) / M=r+8 (lanes 16-31), N=lane%16.
// 16500 tiles = 4125 blocks x 4 waves exactly -> EXEC all-1s everywhere.
// ======================================================================
typedef __attribute__((ext_vector_type(2))) float v2f;
typedef __attribute__((ext_vector_type(8))) float v8f;

__global__ void ppc_reduce_wmma(const float* __restrict__ frame,
                                const float* __restrict__ ind,
                                float* __restrict__ off_out,
                                float* __restrict__ def_out) {
    int gtid = blockIdx.x * blockDim.x + threadIdx.x;
    int wave = gtid >> 5;
    int lane = gtid & 31;
    int rowBase = wave * 16;               // 16 (f,jf) rows per tile
    int m  = lane & 15;
    int kh = lane >> 4;

    const float* arow = ind + (size_t)(rowBase + m) * PC;
    v8f c = {};
    #pragma unroll
    for (int kc = 0; kc < 6; ++kc) {       // K = 24 (22 players + 2 zero pad)
        int k0 = 4 * kc + 2 * kh;
        v2f a, b;
        a[0] = (k0     < PC) ? arow[k0]     : 0.0f;
        a[1] = (k0 + 1 < PC) ? arow[k0 + 1] : 0.0f;
        float t0 = (k0     < PC) ? frame[k0 * 14 + 7]       : 0.0f;
        float t1 = (k0 + 1 < PC) ? frame[(k0 + 1) * 14 + 7] : 0.0f;
        // column 0 = offense weight, column 1 = defense weight
        b[0] = (m == 0) ? t0 : ((m == 1 && k0     < PC) ? 1.0f - t0 : 0.0f);
        b[1] = (m == 0) ? t1 : ((m == 1 && k0 + 1 < PC) ? 1.0f - t1 : 0.0f);
        c = __builtin_amdgcn_wmma_f32_16x16x4_f32(
                /*neg_a=*/false, a, /*neg_b=*/false, b,
                /*c_mod=*/(short)0, c, /*reuse_a=*/false, /*reuse_b=*/false);
    }

    int rowOff = rowBase + kh * 8;
    if (m == 0) {
        #pragma unroll
        for (int r = 0; r < 8; ++r) off_out[rowOff + r] = c[r];
    } else if (m == 1) {
        #pragma unroll
        for (int r = 0; r < 8; ++r) def_out[rowOff + r] = c[r];
    }
}

// ======================================================================
extern "C" void kernel_launch(void* const* d_in, const int* in_sizes, int n_in,
                              void* d_out, int out_size, void* d_ws, size_t ws_size,
                              hipStream_t stream) {
    const float* frame = (const float*)d_in[0];       // (1,22,14) fp32
    float* table   = (float*)d_ws;                    // F*NT*P fp32 = 23.2 MB
    float* off_out = (float*)d_out;                   // F*NT
    float* def_out = off_out + (size_t)FC * NTC;      // F*NT
    float* ind_out = def_out + (size_t)FC * NTC;      // F*NT*P

    // 1) p_int table (145200 threads)
    pint_kernel<<<(FC * PC + 255) / 256, 256, 0, stream>>>(frame, table);

    // 2) path walk: 264000 waves = 33000 blocks x 8 waves (exact)
    path_kernel<<<(FC * NTC * 32) / 256, 256, 0, stream>>>(frame, table, ind_out);

    // 3) WMMA team reduction: 16500 tiles = 4125 blocks x 4 waves (exact)
    ppc_reduce_wmma<<<4125, 128, 0, stream>>>(frame, ind_out, off_out, def_out);
}